// GraphSAGE_44220983280302
// MI455X (gfx1250) — compile-verified
//
#include <hip/hip_runtime.h>
#include <hip/hip_bf16.h>
#include <stdint.h>

// GraphSAGE (2-layer, mean aggregation) for MI455X / gfx1250.
//   h   = relu( (scatter_mean(x, src->dst)) @ W1_l + b1 + x @ W1_r )
//   out =       (scatter_mean(h, src->dst)) @ W2_l + b2 + h @ W2_r
//
// Roofline: edge scatter dominates (~1.6 GB -> ~70us @ 23.3 TB/s HBM);
// GEMMs are ~5 GFLOP, done in full f32 precision with V_WMMA_F32_16X16X4_F32.
// Tile staging uses GLOBAL_LOAD_ASYNC_TO_LDS_B128 (ASYNCcnt path), scatter
// uses native global_atomic_add_f32 (no-return, device scope).

#define N_NODES 50000
#define N_EDGES 800000
#define IN_DIM  128
#define HID_DIM 128
#define OUT_DIM 64

typedef __attribute__((ext_vector_type(2))) float v2f;
typedef __attribute__((ext_vector_type(8))) float v8f;

// Native f32 atomic add, no return value (STOREcnt-tracked), device scope.
__device__ __forceinline__ void gatomic_add_f32(float* p, float v) {
    asm volatile("global_atomic_add_f32 %0, %1, off scope:SCOPE_DEV"
                 :: "v"(p), "v"(v) : "memory");
}

// gfx1250 async global->LDS copy, 16B per lane, ASYNCcnt-tracked.
__device__ __forceinline__ void async_copy_b128(uint32_t lds_off, const void* g) {
    asm volatile("global_load_async_to_lds_b128 %0, %1, off"
                 :: "v"(lds_off), "v"(g) : "memory");
}

__device__ __forceinline__ void wait_asynccnt0() {
    asm volatile("s_wait_asynccnt 0" ::: "memory");
}

// ---------------------------------------------------------------------------
// Edge scatter: sum[dst] += feat[src] (+ deg[dst] += 1 on first layer).
// 32 threads per edge, one float4 chunk each -> coalesced 512B gather/edge.
// ---------------------------------------------------------------------------
template <int D, bool DO_DEG>
__global__ __launch_bounds__(256)
void scatter_add_kernel(const float* __restrict__ feat,
                        const int* __restrict__ eidx,
                        float* __restrict__ sum,
                        float* __restrict__ deg) {
    constexpr int CH = D / 4;  // float4 chunks per edge (32 for D=128)
    long long gid = (long long)blockIdx.x * blockDim.x + threadIdx.x;
    if (gid >= (long long)N_EDGES * CH) return;
    int e = (int)(gid / CH);
    int c = (int)(gid - (long long)e * CH);
    int src = eidx[e];
    int dst = eidx[N_EDGES + e];

    if (DO_DEG && c == 0) {
        gatomic_add_f32(&deg[dst], 1.0f);
    }
    const float4 v = reinterpret_cast<const float4*>(feat)[(size_t)src * CH + c];
    float* dp = sum + (size_t)dst * D + c * 4;
    gatomic_add_f32(dp + 0, v.x);
    gatomic_add_f32(dp + 1, v.y);
    gatomic_add_f32(dp + 2, v.z);
    gatomic_add_f32(dp + 3, v.w);
}

// ---------------------------------------------------------------------------
// Fused SAGE layer: out = act( (sum/deg) @ Wl + bias + x @ Wr )
// One block = 16 nodes x OUT features; one wave32 per 16x16 tile using
// V_WMMA_F32_16X16X4_F32, K-loop over IN in steps of 4.
//
// A-operand (32-bit A 16x4, ISA 7.12.2): lane -> row = lane&15,
//   K-pair base = (lane>>4)*2; .x = K, .y = K+1 (ds_load_b64-able from LDS).
// B-operand (4x16) mirrors that with N = lane&15.
// C/D: VGPR r -> row = r + 8*(lane>=16), col = lane&15.
// Degree normalization: 1/max(deg,1) computed ONCE per node into LDS
// (16 IEEE divides per block), then applied as a row-scale of the
// accumulator between the two GEMM loops (linear in rows, so equivalent
// to scaling the aggregated input).
// ---------------------------------------------------------------------------
template <int IN, int OUT, bool RELU>
__global__ __launch_bounds__(32 * (OUT / 16))
void sage_layer_kernel(const float* __restrict__ xin,
                       const float* __restrict__ sum,
                       const float* __restrict__ deg,
                       const float* __restrict__ Wl,
                       const float* __restrict__ Wr,
                       const float* __restrict__ bias,
                       float* __restrict__ outp) {
    constexpr int NT  = 32 * (OUT / 16);
    constexpr int PAD = 4;                  // row pad -> conflict-free A reads
    __shared__ float sAgg[16][IN + PAD];    // raw per-dst sums (unscaled)
    __shared__ float sX[16][IN + PAD];
    __shared__ float sInv[16];              // per-node 1/max(deg,1)

    const int m0  = blockIdx.x * 16;        // node tile base (N_NODES % 16 == 0)
    const int tid = threadIdx.x;

    // Async-stage x-tile and sum-tile into LDS (16B per lane, ASYNCcnt).
    for (int idx = tid; idx < 16 * (IN / 4); idx += NT) {
        int row = idx / (IN / 4);
        int c4  = idx - row * (IN / 4);
        int node = m0 + row;
        const float* gx = xin + (size_t)node * IN + c4 * 4;
        const float* gs = sum + (size_t)node * IN + c4 * 4;
        uint32_t lx = (uint32_t)(uintptr_t)&sX[row][c4 * 4];
        uint32_t ls = (uint32_t)(uintptr_t)&sAgg[row][c4 * 4];
        async_copy_b128(lx, gx);
        async_copy_b128(ls, gs);
    }
    // One IEEE divide per node (not per lane) for the mean normalization.
    if (tid < 16) {
        sInv[tid] = 1.0f / fmaxf(deg[m0 + tid], 1.0f);
    }
    wait_asynccnt0();       // our own async writes to LDS complete
    __syncthreads();        // everyone's writes visible to everyone

    const int lane = tid & 31;
    const int wave = tid >> 5;
    const int n0   = wave * 16;     // output-column tile base
    const int l15  = lane & 15;     // A row / B col within tile
    const int half = lane >> 4;     // K-half selector

    v8f acc = {};

    // sum @ Wl  (unscaled)
    #pragma unroll 4
    for (int k = 0; k < IN; k += 4) {
        int kk = k + half * 2;
        v2f a; a.x = sAgg[l15][kk];            a.y = sAgg[l15][kk + 1];
        v2f b; b.x = Wl[kk * OUT + n0 + l15];  b.y = Wl[(kk + 1) * OUT + n0 + l15];
        acc = __builtin_amdgcn_wmma_f32_16x16x4_f32(false, a, false, b,
                                                    (short)0, acc, false, false);
    }

    // Row-scale accumulator by 1/deg (broadcast ds_load + v_mul per row).
    #pragma unroll
    for (int r = 0; r < 8; ++r) {
        acc[r] *= sInv[half * 8 + r];
    }

    // + x @ Wr (accumulates into same C)
    #pragma unroll 4
    for (int k = 0; k < IN; k += 4) {
        int kk = k + half * 2;
        v2f a; a.x = sX[l15][kk];              a.y = sX[l15][kk + 1];
        v2f b; b.x = Wr[kk * OUT + n0 + l15];  b.y = Wr[(kk + 1) * OUT + n0 + l15];
        acc = __builtin_amdgcn_wmma_f32_16x16x4_f32(false, a, false, b,
                                                    (short)0, acc, false, false);
    }

    const float bv = bias[n0 + l15];
    #pragma unroll
    for (int r = 0; r < 8; ++r) {
        float v = acc[r] + bv;
        if (RELU) v = fmaxf(v, 0.0f);
        int row = m0 + r + half * 8;
        outp[(size_t)row * OUT + n0 + l15] = v;
    }
}

// ---------------------------------------------------------------------------
extern "C" void kernel_launch(void* const* d_in, const int* in_sizes, int n_in,
                              void* d_out, int out_size, void* d_ws, size_t ws_size,
                              hipStream_t stream) {
    (void)in_sizes; (void)n_in; (void)out_size; (void)ws_size;

    const float* x   = (const float*)d_in[0];
    const int*   ei  = (const int*)d_in[1];   // [2, E]: row0 = src, row1 = dst
    const float* W1l = (const float*)d_in[2];
    const float* W1r = (const float*)d_in[3];
    const float* b1  = (const float*)d_in[4];
    const float* W2l = (const float*)d_in[5];
    const float* W2r = (const float*)d_in[6];
    const float* b2  = (const float*)d_in[7];
    float* out = (float*)d_out;

    // Workspace carve-out (256B aligned): deg | sum1 | h | sum2  (~77 MB)
    char* ws = (char*)d_ws;
    size_t off = 0;
    auto carve = [&](size_t bytes) -> float* {
        float* p = (float*)(ws + off);
        off += (bytes + 255) & ~(size_t)255;
        return p;
    };
    float* deg  = carve((size_t)N_NODES * 4);
    float* sum1 = carve((size_t)N_NODES * IN_DIM * 4);
    float* h    = carve((size_t)N_NODES * HID_DIM * 4);
    float* sum2 = carve((size_t)N_NODES * HID_DIM * 4);

    hipMemsetAsync(deg,  0, (size_t)N_NODES * 4, stream);
    hipMemsetAsync(sum1, 0, (size_t)N_NODES * IN_DIM * 4, stream);
    hipMemsetAsync(sum2, 0, (size_t)N_NODES * HID_DIM * 4, stream);

    // Layer 1: scatter-mean of x, then fused GEMM+bias+ReLU -> h
    {
        long long total = (long long)N_EDGES * (IN_DIM / 4);
        int blocks = (int)((total + 255) / 256);
        scatter_add_kernel<IN_DIM, true><<<blocks, 256, 0, stream>>>(x, ei, sum1, deg);
    }
    sage_layer_kernel<IN_DIM, HID_DIM, true>
        <<<N_NODES / 16, 32 * (HID_DIM / 16), 0, stream>>>(x, sum1, deg, W1l, W1r, b1, h);

    // Layer 2: scatter-mean of h, then fused GEMM+bias -> out
    {
        long long total = (long long)N_EDGES * (HID_DIM / 4);
        int blocks = (int)((total + 255) / 256);
        scatter_add_kernel<HID_DIM, false><<<blocks, 256, 0, stream>>>(h, ei, sum2, nullptr);
    }
    sage_layer_kernel<HID_DIM, OUT_DIM, false>
        <<<N_NODES / 16, 32 * (OUT_DIM / 16), 0, stream>>>(h, sum2, deg, W2l, W2r, b2, out);
}